// Encoder_66520453480545
// MI455X (gfx1250) — compile-verified
//
#include <hip/hip_runtime.h>

// GCN forward (second conv only; conv1 is dead in the reference output).
// z = D^-1/2 (A+I) D^-1/2 X W2 + b2
//
// Pipeline:
//   1) gcn_gemm_wmma : xw = X @ W2 via V_WMMA_F32_16X16X4_F32 (f32-exact)
//   2) deg_init      : deg[i] = 1.0 (self loop)
//   3) deg_edges     : deg[dst[e]] += 1 (float atomics; integer-valued, exact)
//   4) deg_rsqrt     : isd[i] = rsqrt(deg[i])   (deg >= 1 always)
//   5) out_init      : out[i,c] = b2[c] + isd[i]^2 * xw[i,c]  (self-loop term)
//   6) edge_scatter  : out[dst,c] += isd[src]*isd[dst]*xw[src,c]  (64 lanes/edge)

typedef float v2f __attribute__((ext_vector_type(2)));
typedef float v8f __attribute__((ext_vector_type(8)));

#define IN_CH  128
#define OUT_CH 64

// ---------------------------------------------------------------------------
// 1) xw = X[N,128] @ W[128,64], one 16-row tile per block, one 16-col tile
//    per wave (4 waves/block). N % 16 == 0 (50000 = 3125*16), so every wave
//    is fully active: EXEC is all-ones as WMMA requires.
// ---------------------------------------------------------------------------
__global__ __launch_bounds__(128) void gcn_gemm_wmma(const float* __restrict__ X,
                                                     const float* __restrict__ W,
                                                     float* __restrict__ XW) {
    __shared__ float Wlds[IN_CH * OUT_CH];             // 32 KB of the 320 KB WGP LDS
    for (int i = threadIdx.x; i < IN_CH * OUT_CH; i += 128) Wlds[i] = W[i];
    __syncthreads();

    const int lane = threadIdx.x & 31;
    const int wave = threadIdx.x >> 5;                 // column tile 0..3
    const int m    = lane & 15;
    const int hi   = lane >> 4;                        // half-wave select
    const int row0 = blockIdx.x * 16;

    // A-fragment source: row (row0+m), K pairs {4t+2hi, 4t+2hi+1}
    const v2f* xrow = (const v2f*)(X + (size_t)(row0 + m) * IN_CH);
    const int col = wave * 16 + m;                     // B/C/D column (N index)

    v8f acc = {};
#pragma unroll
    for (int t = 0; t < IN_CH / 4; ++t) {
        v2f a = xrow[2 * t + hi];                      // A 16x4 f32 layout (ISA 7.12.2)
        const int kb = 4 * t + 2 * hi;
        v2f b;
        b.x = Wlds[(kb + 0) * OUT_CH + col];           // B 4x16: row K striped across lanes
        b.y = Wlds[(kb + 1) * OUT_CH + col];
        // D = A*B + C ; 8 args: (neg_a, A, neg_b, B, c_mod, C, reuse_a, reuse_b)
        acc = __builtin_amdgcn_wmma_f32_16x16x4_f32(false, a, false, b,
                                                    (short)0, acc, false, false);
    }

    // C/D layout: VGPR r holds row (r + 8*hi), column = lane&15 within tile.
    float* outp = XW + (size_t)row0 * OUT_CH + wave * 16 + m;
#pragma unroll
    for (int r = 0; r < 8; ++r)
        outp[(size_t)(r + hi * 8) * OUT_CH] = acc[r];
}

// ---------------------------------------------------------------------------
// 2-4) degree and inverse-sqrt-degree
// ---------------------------------------------------------------------------
__global__ void deg_init(float* __restrict__ deg, int n) {
    int i = blockIdx.x * blockDim.x + threadIdx.x;
    if (i < n) deg[i] = 1.0f;                          // self loop
}

__global__ void deg_edges(const int* __restrict__ dst, float* __restrict__ deg, int nE) {
    int e = blockIdx.x * blockDim.x + threadIdx.x;
    if (e < nE) atomicAdd(&deg[dst[e]], 1.0f);
}

__global__ void deg_rsqrt(const float* __restrict__ deg, float* __restrict__ isd, int n) {
    int i = blockIdx.x * blockDim.x + threadIdx.x;
    if (i < n) isd[i] = rsqrtf(deg[i]);
}

// ---------------------------------------------------------------------------
// 5) out[i,c] = b2[c] + isd[i]^2 * xw[i,c]
// ---------------------------------------------------------------------------
__global__ void out_init(const float* __restrict__ xw, const float* __restrict__ isd,
                         const float* __restrict__ b2, float* __restrict__ out, int total) {
    int i = blockIdx.x * blockDim.x + threadIdx.x;
    if (i >= total) return;
    int node = i >> 6;
    int ch   = i & 63;
    float s = isd[node];
    out[i] = b2[ch] + s * s * xw[i];
}

// ---------------------------------------------------------------------------
// 6) edge scatter: 64 consecutive threads handle one edge's 64 channels so the
//    gather of xw[src,:] (256 B) and the atomics on out[dst,:] are coalesced.
// ---------------------------------------------------------------------------
__global__ __launch_bounds__(256) void edge_scatter(const int* __restrict__ src,
                                                    const int* __restrict__ dst,
                                                    const float* __restrict__ xw,
                                                    const float* __restrict__ isd,
                                                    float* __restrict__ out, int nE) {
    int e  = blockIdx.x * (blockDim.x >> 6) + (threadIdx.x >> 6);
    int ch = threadIdx.x & 63;
    if (e >= nE) return;
    int s = src[e];
    int d = dst[e];
    float c = isd[s] * isd[d];
    atomicAdd(&out[(size_t)d * OUT_CH + ch], c * xw[(size_t)s * OUT_CH + ch]);
}

// ---------------------------------------------------------------------------
extern "C" void kernel_launch(void* const* d_in, const int* in_sizes, int n_in,
                              void* d_out, int out_size, void* d_ws, size_t ws_size,
                              hipStream_t stream) {
    const int*   edges = (const int*)d_in[0];      // [2, E] int32
    const float* X     = (const float*)d_in[1];    // [N, 128]
    const float* W2    = (const float*)d_in[4];    // [128, 64]
    const float* b2    = (const float*)d_in[5];    // [64]
    float*       out   = (float*)d_out;            // [N, 64]

    const int E = in_sizes[0] / 2;                 // 800000
    const int N = in_sizes[1] / IN_CH;             // 50000

    const int* src = edges;
    const int* dst = edges + E;

    float* xw  = (float*)d_ws;                     // N*64 floats
    float* deg = xw + (size_t)N * OUT_CH;          // N floats
    float* isd = deg + N;                          // N floats

    // 1) GEMM (WMMA): one 16-row tile per block, 4 col-tile waves per block
    gcn_gemm_wmma<<<N / 16, 128, 0, stream>>>(X, W2, xw);

    // 2-4) degrees
    deg_init <<<(N + 255) / 256, 256, 0, stream>>>(deg, N);
    deg_edges<<<(E + 255) / 256, 256, 0, stream>>>(dst, deg, E);
    deg_rsqrt<<<(N + 255) / 256, 256, 0, stream>>>(deg, isd, N);

    // 5) bias + self-loop term
    out_init<<<((N * OUT_CH) + 255) / 256, 256, 0, stream>>>(xw, isd, b2, out, N * OUT_CH);

    // 6) edge aggregation (4 edges per 256-thread block)
    edge_scatter<<<(E + 3) / 4, 256, 0, stream>>>(src, dst, xw, isd, out, E);
}